// NBMEHead_7310034337830
// MI455X (gfx1250) — compile-verified
//
#include <hip/hip_runtime.h>

typedef __attribute__((ext_vector_type(2))) float v2f;
typedef __attribute__((ext_vector_type(8))) float v8f;

#define D_DIM 1024
#define O_DIM 3
#define WROWS 4    // W staged with a zero 4th row so lanes n>=3 read zeros (no mask mul)
#define BK 32
#define ROWS_PER_WG 128
#define PITCH 36   // floats; stride 36 banks -> conflict-free b64 LDS reads

// One-round counter-based mixer for the per-element dropout masks.
__device__ __forceinline__ unsigned hmix(unsigned v) {
  v ^= v >> 16; v *= 0x7feb352dU;
  v ^= v >> 15; v *= 0x846ca68bU;
  v ^= v >> 16;
  return v;
}

// mean over 5 inverted-dropout passes, p = 0.1..0.5:
// m = (1/5) * sum_s keep_s / (1 - p_s), keep_s ~ Bernoulli(1 - p_s).
// All 5 masks come from independent 6-bit fields of a single 32-bit hash
// (keep-prob resolution 1/64; p=0.5 exact).
__device__ __forceinline__ float drop_avg_scale(unsigned e) {
  const unsigned h = hmix(e * 0x9E3779B9U + 0x01234567U);
  float m;
  m  = ((h         & 63u) >= 6u)  ? 0.22222222f : 0.0f;  // p=0.1 -> 1/(5*0.9)
  m += (((h >> 6)  & 63u) >= 13u) ? 0.25f       : 0.0f;  // p=0.2 -> 1/(5*0.8)
  m += (((h >> 12) & 63u) >= 19u) ? 0.28571429f : 0.0f;  // p=0.3 -> 1/(5*0.7)
  m += (((h >> 18) & 63u) >= 26u) ? 0.33333333f : 0.0f;  // p=0.4 -> 1/(5*0.6)
  m += (((h >> 24) & 63u) >= 32u) ? 0.40f       : 0.0f;  // p=0.5 -> 1/(5*0.5)
  return m;
}

__global__ __launch_bounds__(256) void nbme_head_kernel(
    const float* __restrict__ x, const float* __restrict__ W,
    const float* __restrict__ b, float* __restrict__ out) {
  __shared__ float ldsA[ROWS_PER_WG * PITCH];
  __shared__ float ldsW[WROWS * D_DIM];

  const int tid  = threadIdx.x;
  const int wave = tid >> 5;
  const int lane = tid & 31;
  const int rowBase = blockIdx.x * ROWS_PER_WG;

  // Stage W [3 x 1024] in LDS, 4th row zeroed (first loop barrier covers this).
  for (int i = tid; i < WROWS * D_DIM; i += 256)
    ldsW[i] = (i < O_DIM * D_DIM) ? W[i] : 0.0f;

  // WMMA lane roles (16x16x4 f32): lane L holds A row M = L&15, K = 2*(L>>4)+{0,1}
  const int n  = lane & 15;                   // B/C column index for this lane
  const int kh = lane >> 4;                   // which K-half this lane holds
  const int aRow = wave * 16 + n;             // A-fragment row in the LDS tile
  const int nIdx = (n < O_DIM) ? n : O_DIM;   // lanes n>=3 read the zero row

  v8f acc = {};

  for (int kb = 0; kb < D_DIM; kb += BK) {
    __syncthreads();  // previous tile fully consumed

    // Cooperative 128x32 tile load, dropout-average fused in.
    #pragma unroll
    for (int i = 0; i < 4; ++i) {
      const int f = tid + i * 256;       // float4 slot 0..1023
      const int r = f >> 3;              // tile row 0..127
      const int c = (f & 7) << 2;        // tile col 0..28 step 4
      const unsigned g = (unsigned)(rowBase + r) * D_DIM + (unsigned)(kb + c);
      const float4 v = *reinterpret_cast<const float4*>(x + g);
      float4 s;
      s.x = v.x * drop_avg_scale(g + 0u);
      s.y = v.y * drop_avg_scale(g + 1u);
      s.z = v.z * drop_avg_scale(g + 2u);
      s.w = v.w * drop_avg_scale(g + 3u);
      *reinterpret_cast<float4*>(&ldsA[r * PITCH + c]) = s;
      if (kb + BK < D_DIM) {
        __builtin_prefetch(x + g + BK, 0, 1);  // global_prefetch_b8 for next tile
      }
    }
    __syncthreads();

    #pragma unroll
    for (int k4 = 0; k4 < BK; k4 += 4) {
      // A fragment: two consecutive f32 at (row, k4 + 2*kh) — conflict-free ds_load_b64
      const v2f a = *reinterpret_cast<const v2f*>(&ldsA[aRow * PITCH + k4 + 2 * kh]);
      // B fragment: W^T column n (row 3 of ldsW is zeros for lanes n>=3)
      const int kcol = kb + k4 + 2 * kh;
      v2f bb;
      bb.x = ldsW[nIdx * D_DIM + kcol + 0];
      bb.y = ldsW[nIdx * D_DIM + kcol + 1];
      // D = A(16x4) * B(4x16) + C, f32 accumulate
      acc = __builtin_amdgcn_wmma_f32_16x16x4_f32(
          false, a, false, bb, (short)0, acc, false, false);
    }
  }

  // Epilogue: C layout -> VGPR g holds (M = g | 8+g, N = lane&15). Add bias, store N<3.
  if (n < O_DIM) {
    const float bias = b[n];
    #pragma unroll
    for (int g = 0; g < 8; ++g) {
      const int m   = (lane < 16) ? g : (8 + g);
      const int row = rowBase + wave * 16 + m;
      out[row * O_DIM + n] = acc[g] + bias;
    }
  }
}

extern "C" void kernel_launch(void* const* d_in, const int* in_sizes, int n_in,
                              void* d_out, int out_size, void* d_ws, size_t ws_size,
                              hipStream_t stream) {
  const float* x = (const float*)d_in[0];
  const float* W = (const float*)d_in[1];
  const float* b = (const float*)d_in[2];
  float* out = (float*)d_out;

  const int M = in_sizes[0] / D_DIM;           // 32*2048 = 65536 rows
  dim3 grid(M / ROWS_PER_WG);                  // 512 workgroups
  dim3 block(256);                             // 8 wave32s
  hipLaunchKernelGGL(nbme_head_kernel, grid, block, 0, stream, x, W, b, out);
}